// MultiHeadAttention_2791728742637
// MI455X (gfx1250) — compile-verified
//
#include <hip/hip_runtime.h>

// ---------------------------------------------------------------------------
// MHA forward for MI455X (gfx1250, wave32, WMMA).
// Precision: fp32 throughout via V_WMMA_F32_16X16X4_F32 (reference is fp32 and
// the problem is HBM-bound on the 512MB attn write, so low-precision WMMA
// buys nothing).
// ---------------------------------------------------------------------------

typedef __attribute__((ext_vector_type(2))) float v2f;
typedef __attribute__((ext_vector_type(8))) float v8f;

#define D_MODEL 1024
#define SEQ     2048
#define BATCH   2
#define NHEAD   16
#define DEPTH   64
#define ROWS    (BATCH * SEQ)     // 4096 rows for the projection GEMMs
#define LDSP    68                // padded LDS row stride (floats): 64+4 kills
                                  // the 64-bank stride-64 conflict

__device__ __forceinline__ v8f wmma4(v2f a, v2f b, v8f c) {
  // D = A(16x4,f32) * B(4x16,f32) + C(16x16,f32)
  return __builtin_amdgcn_wmma_f32_16x16x4_f32(
      /*neg_a=*/false, a, /*neg_b=*/false, b,
      /*c_mod=*/(short)0, c, /*reuse_a=*/false, /*reuse_b=*/false);
}

// ---------------------------------------------------------------------------
// Tiled GEMM + bias: Y[M,N] = X[M,K] @ W[K,N] + bias[N]
// Block = 128 threads (4 waves). Block tile 64x64, K-chunk 64.
// Wave w computes rows [16w,16w+16) x all 64 cols (4 col-tiles of 16).
// ---------------------------------------------------------------------------
__global__ __launch_bounds__(128)
void gemm_bias_wmma(const float* __restrict__ X, const float* __restrict__ W,
                    const float* __restrict__ bias, float* __restrict__ Y,
                    int M, int N, int K) {
  __shared__ float Xs[64][LDSP];
  __shared__ float Ws[64][LDSP];

  const int tid  = threadIdx.x;
  const int wave = tid >> 5;
  const int lane = tid & 31;
  const int half = lane >> 4;   // which 16-lane half
  const int l16  = lane & 15;
  const int m0   = blockIdx.x * 64;
  const int n0   = blockIdx.y * 64;

  v8f acc[4];
#pragma unroll
  for (int t = 0; t < 4; ++t)
#pragma unroll
    for (int e = 0; e < 8; ++e) acc[t][e] = 0.0f;

  const int myrow = wave * 16 + l16;      // A-frag row: M = lane%16

#pragma unroll 1
  for (int kc = 0; kc < K; kc += 64) {
    // cooperative float4 loads: 64x64 block = 1024 float4, 128 threads
#pragma unroll
    for (int i = tid; i < 64 * 16; i += 128) {
      const int r = i >> 4, c4 = i & 15;
      *(float4*)&Xs[r][c4 * 4] =
          *(const float4*)(X + (size_t)(m0 + r) * K + kc + c4 * 4);
      *(float4*)&Ws[r][c4 * 4] =
          *(const float4*)(W + (size_t)(kc + r) * N + n0 + c4 * 4);
    }
    __syncthreads();

#pragma unroll
    for (int kk = 0; kk < 16; ++kk) {
      const int k0 = kk * 4 + half * 2;   // K = 2*(lane/16) + {0,1}
      v2f a;
      a.x = Xs[myrow][k0];
      a.y = Xs[myrow][k0 + 1];
#pragma unroll
      for (int ct = 0; ct < 4; ++ct) {
        v2f b;
        b.x = Ws[k0][ct * 16 + l16];
        b.y = Ws[k0 + 1][ct * 16 + l16];
        acc[ct] = wmma4(a, b, acc[ct]);
      }
    }
    __syncthreads();
  }

  // epilogue: bias + store (C/D layout: row = r + 8*half, col = lane%16)
#pragma unroll
  for (int ct = 0; ct < 4; ++ct) {
    const int n = n0 + ct * 16 + l16;
    const float bv = bias[n];
#pragma unroll
    for (int r = 0; r < 8; ++r) {
      const int m = m0 + wave * 16 + r + half * 8;
      Y[(size_t)m * N + n] = acc[ct][r] + bv;
    }
  }
}

// ---------------------------------------------------------------------------
// Attention: one workgroup per (b, h, 64-row q-block); sees full 2048-wide rows.
// Phase 1: logits = scale*Q.K^T via WMMA; online row max/sum in registers;
//          stream raw logits to attn (L2-resident for phase 2 re-read).
// Phase 2: re-read logits, p = exp(x-m)/l; nontemporal-store final attn;
//          ctx += P @ V via WMMA from LDS.
// ---------------------------------------------------------------------------
__global__ __launch_bounds__(128)
void attention_wmma(const float* __restrict__ Qb, const float* __restrict__ Kb,
                    const float* __restrict__ Vb, float* __restrict__ attn,
                    float* __restrict__ Ctx) {
  __shared__ float AS[64][LDSP];   // Q block (phase 1) / P block (phase 2)
  __shared__ float BS[64][LDSP];   // K block (phase 1) / V block (phase 2)

  const int tid  = threadIdx.x;
  const int wave = tid >> 5;
  const int lane = tid & 31;
  const int half = lane >> 4;
  const int l16  = lane & 15;

  const int qblk = blockIdx.x;           // 0..31
  const int bh   = blockIdx.y;           // 0..31
  const int b    = bh >> 4;
  const int h    = bh & 15;
  const size_t hoff  = (size_t)h * DEPTH;
  const size_t qrow0 = (size_t)b * SEQ + qblk * 64;  // row into [B*S, D] bufs
  float* attnP = attn + ((size_t)bh * SEQ + qblk * 64) * SEQ;
  const float scale = 0.125f;            // 1/sqrt(64)

  // load Q block (64 rows x 64 depth)
#pragma unroll
  for (int i = tid; i < 64 * 16; i += 128) {
    const int r = i >> 4, c4 = i & 15;
    *(float4*)&AS[r][c4 * 4] =
        *(const float4*)(Qb + (qrow0 + r) * D_MODEL + hoff + c4 * 4);
  }
  __syncthreads();

  float mrow[8], lrow[8];                // stats for rows r + 8*half
#pragma unroll
  for (int r = 0; r < 8; ++r) { mrow[r] = -3.0e38f; lrow[r] = 0.0f; }

  const int myrow = wave * 16 + l16;

  // ---------------- Phase 1: logits + online softmax stats ----------------
#pragma unroll 1
  for (int kb = 0; kb < SEQ / 64; ++kb) {
    const size_t krow0 = (size_t)b * SEQ + kb * 64;
#pragma unroll
    for (int i = tid; i < 64 * 16; i += 128) {
      const int r = i >> 4, c4 = i & 15;
      *(float4*)&BS[r][c4 * 4] =
          *(const float4*)(Kb + (krow0 + r) * D_MODEL + hoff + c4 * 4);
    }
    __syncthreads();

    v8f lg[4];
#pragma unroll
    for (int t = 0; t < 4; ++t)
#pragma unroll
      for (int e = 0; e < 8; ++e) lg[t][e] = 0.0f;

#pragma unroll
    for (int kk = 0; kk < 16; ++kk) {
      const int k0 = kk * 4 + half * 2;
      v2f a;
      a.x = AS[myrow][k0];
      a.y = AS[myrow][k0 + 1];
#pragma unroll
      for (int ct = 0; ct < 4; ++ct) {
        // logits[m][n] = sum_d Q[m][d]*K[n][d]: B-frag rows come from K rows n
        v2f bf;
        bf.x = BS[ct * 16 + l16][k0];
        bf.y = BS[ct * 16 + l16][k0 + 1];
        lg[ct] = wmma4(a, bf, lg[ct]);
      }
    }

    // per-row online max/sum + stream raw logits
#pragma unroll
    for (int r = 0; r < 8; ++r) {
      float vals[4];
      float rmax = -3.0e38f;
#pragma unroll
      for (int ct = 0; ct < 4; ++ct) {
        vals[ct] = lg[ct][r] * scale;
        rmax = fmaxf(rmax, vals[ct]);
      }
#pragma unroll
      for (int off = 1; off < 16; off <<= 1)
        rmax = fmaxf(rmax, __shfl_xor(rmax, off, 32));   // stays in 16-half
      const float newm = fmaxf(mrow[r], rmax);
      float s = 0.0f;
#pragma unroll
      for (int ct = 0; ct < 4; ++ct) s += __expf(vals[ct] - newm);
#pragma unroll
      for (int off = 1; off < 16; off <<= 1) s += __shfl_xor(s, off, 32);
      lrow[r] = lrow[r] * __expf(mrow[r] - newm) + s;
      mrow[r] = newm;

      const int grow = wave * 16 + r + half * 8;
#pragma unroll
      for (int ct = 0; ct < 4; ++ct)
        attnP[(size_t)grow * SEQ + kb * 64 + ct * 16 + l16] = vals[ct];
    }
    __syncthreads();
  }

  // ---------------- Phase 2: normalize + ctx = P @ V ----------------
  float invl[8];
#pragma unroll
  for (int r = 0; r < 8; ++r) invl[r] = 1.0f / lrow[r];

  v8f ctx[4];
#pragma unroll
  for (int t = 0; t < 4; ++t)
#pragma unroll
    for (int e = 0; e < 8; ++e) ctx[t][e] = 0.0f;

#pragma unroll 1
  for (int kb = 0; kb < SEQ / 64; ++kb) {
    const size_t krow0 = (size_t)b * SEQ + kb * 64;
#pragma unroll
    for (int i = tid; i < 64 * 16; i += 128) {
      const int r = i >> 4, c4 = i & 15;
      *(float4*)&BS[r][c4 * 4] =
          *(const float4*)(Vb + (krow0 + r) * D_MODEL + hoff + c4 * 4);
    }
    __syncthreads();

    // re-read own raw logits (L2 hit), normalize, NT-store final attn,
    // park P rows in this wave's private stripe of AS (no barrier needed:
    // each wave reads back only the rows it wrote).
#pragma unroll
    for (int r = 0; r < 8; ++r) {
      const int grow = wave * 16 + r + half * 8;
#pragma unroll
      for (int ct = 0; ct < 4; ++ct) {
        const size_t gi = (size_t)grow * SEQ + kb * 64 + ct * 16 + l16;
        const float p = __expf(attnP[gi] - mrow[r]) * invl[r];
        __builtin_nontemporal_store(p, &attnP[gi]);  // never re-read
        AS[grow][ct * 16 + l16] = p;
      }
    }

#pragma unroll
    for (int kk = 0; kk < 16; ++kk) {
      const int k0 = kk * 4 + half * 2;
      v2f a;
      a.x = AS[myrow][k0];
      a.y = AS[myrow][k0 + 1];
#pragma unroll
      for (int ct = 0; ct < 4; ++ct) {
        // ctx[m][d] += sum_n P[m][n] * V[n][d]: B-frag from V rows n = k0..
        v2f bf;
        bf.x = BS[k0][ct * 16 + l16];
        bf.y = BS[k0 + 1][ct * 16 + l16];
        ctx[ct] = wmma4(a, bf, ctx[ct]);
      }
    }
    __syncthreads();
  }

  // write ctx in merged-head layout: Ctx[b*S + s][h*64 + d]
#pragma unroll
  for (int ct = 0; ct < 4; ++ct) {
    const int d = ct * 16 + l16;
#pragma unroll
    for (int r = 0; r < 8; ++r) {
      const int grow = wave * 16 + r + half * 8;
      Ctx[(qrow0 + grow) * D_MODEL + hoff + d] = ctx[ct][r];
    }
  }
}

// ---------------------------------------------------------------------------
extern "C" void kernel_launch(void* const* d_in, const int* in_sizes, int n_in,
                              void* d_out, int out_size, void* d_ws,
                              size_t ws_size, hipStream_t stream) {
  const float* q  = (const float*)d_in[0];
  const float* k  = (const float*)d_in[1];
  const float* v  = (const float*)d_in[2];
  const float* wq = (const float*)d_in[3];
  const float* bq = (const float*)d_in[4];
  const float* wk = (const float*)d_in[5];
  const float* bk = (const float*)d_in[6];
  const float* wv = (const float*)d_in[7];
  const float* bv = (const float*)d_in[8];
  const float* wo = (const float*)d_in[9];
  const float* bo = (const float*)d_in[10];

  float* out  = (float*)d_out;                           // [B,S,D]
  float* attn = out + (size_t)BATCH * SEQ * D_MODEL;     // [B,H,S,S]

  const size_t mat = (size_t)ROWS * D_MODEL;             // 4M floats = 16MB
  float* qbuf = (float*)d_ws;
  float* kbuf = qbuf + mat;
  float* vbuf = kbuf + mat;
  float* cbuf = vbuf + mat;                              // ctx, merged heads

  dim3 blk(128);
  dim3 gproj(ROWS / 64, D_MODEL / 64);                   // 64 x 16 tiles

  gemm_bias_wmma<<<gproj, blk, 0, stream>>>(q, wq, bq, qbuf, ROWS, D_MODEL, D_MODEL);
  gemm_bias_wmma<<<gproj, blk, 0, stream>>>(k, wk, bk, kbuf, ROWS, D_MODEL, D_MODEL);
  gemm_bias_wmma<<<gproj, blk, 0, stream>>>(v, wv, bv, vbuf, ROWS, D_MODEL, D_MODEL);

  dim3 gattn(SEQ / 64, BATCH * NHEAD);                   // 32 x 32
  attention_wmma<<<gattn, blk, 0, stream>>>(qbuf, kbuf, vbuf, attn, cbuf);

  gemm_bias_wmma<<<gproj, blk, 0, stream>>>(cbuf, wo, bo, out, ROWS, D_MODEL, D_MODEL);
}